// v_PTC_5d_84310208021232
// MI455X (gfx1250) — compile-verified
//
#include <hip/hip_runtime.h>
#include <hip/hip_bf16.h>

// PTC-5D (parallel transport convolution) for MI455X / gfx1250.
// complex64 math kept in fp32; spin contraction (16x16 complex) done with
// chained v_wmma_f32_16x16x4_f32; SU(3) color transport done in VALU.

typedef __attribute__((ext_vector_type(2))) float v2f;   // A/B frag: 16x4 / 4x16 f32
typedef __attribute__((ext_vector_type(8))) float v8f;   // C/D frag: 16x16 f32

#define LAT_V          262144   // 8*16*16*16*8
#define FIN            4
#define FOUT           4
#define NPATH          11
#define SITES_PER_WAVE 16
#define WAVES_PER_BLK  8

// lattice (s,a,b,c,d) = (8,16,16,16,8); x bit layout s[17:15] a[14:11] b[10:7] c[6:3] d[2:0]
__device__ __forceinline__ int shift_site(int x, int mu, int d) {
  const int sh[5]  = {15, 11, 7, 3, 0};
  const int msk[5] = {7, 15, 15, 15, 7};
  const int b = sh[mu], m = msk[mu];
  int comp = ((x >> b) + d) & m;
  return (x & ~(m << b)) | (comp << b);
}

__global__ __launch_bounds__(256, 1) void ptc5d_wmma_kernel(
    const float* __restrict__ feat,   // [FIN][V][4 spin][3 color] complex (re,im)
    const float* __restrict__ U,      // [5][V][3][3] complex
    const float* __restrict__ W,      // [FIN][FOUT][NPATH][4][4] complex
    float* __restrict__ out)          // [FOUT][V][4][3] complex
{
  const int lane = threadIdx.x & 31;
  const int wave = blockIdx.x * WAVES_PER_BLK + (threadIdx.x >> 5);
  const int n    = lane & 15;          // column / A-row index within fragments
  const int half = lane >> 4;          // lane group (selects K'/M offset)
  const int x    = wave * SITES_PER_WAVE + n;

  // 6 resident accumulators: D[(o*4+I)][site] for color G=0..2, re/im
  v8f acc_re[3], acc_im[3];
#pragma unroll
  for (int g = 0; g < 3; ++g)
#pragma unroll
    for (int e = 0; e < 8; ++e) { acc_re[g][e] = 0.f; acc_im[g][e] = 0.f; }

  const int oo = n >> 2, II = n & 3;   // A-matrix row M = o*4 + I = n

  for (int p = 0; p < NPATH; ++p) {
    // ---------- gauge-transport setup: source site xn, 3x3 complex matrix M ----------
    int xn = x;
    float Mre[3][3], Mim[3][3];
    const bool ident = (p == 0);
    if (!ident) {
      if (p <= 5) {                 // path (mu,+1): t[x] = U[mu]^dag[x-e] v[x-e]
        const int mu = p - 1;
        xn = shift_site(x, mu, -1);
        const float* u = U + ((size_t)mu * LAT_V + (size_t)xn) * 18;
#pragma unroll
        for (int i = 0; i < 3; ++i)
#pragma unroll
          for (int j = 0; j < 3; ++j) {
            Mre[i][j] =  u[(j * 3 + i) * 2 + 0];   // adjoint
            Mim[i][j] = -u[(j * 3 + i) * 2 + 1];
          }
      } else {                      // path (mu,-1): t[x] = U[mu][x] v[x+e]
        const int mu = p - 6;
        xn = shift_site(x, mu, +1);
        const float* u = U + ((size_t)mu * LAT_V + (size_t)x) * 18;
#pragma unroll
        for (int i = 0; i < 3; ++i)
#pragma unroll
          for (int j = 0; j < 3; ++j) {
            Mre[i][j] = u[(i * 3 + j) * 2 + 0];
            Mim[i][j] = u[(i * 3 + j) * 2 + 1];
          }
      }
    }

    // ---------- build B fragments: transported features t[(f,J)][site][G] ----------
    // B 4x16 layout: VGPR r, lanes 0-15 -> K' = r (half=0) / K' = 2+r (half=1)
    v2f b_re[3][4], b_im[3][4];      // [G][K-chunk]
#pragma unroll
    for (int kc = 0; kc < 4; ++kc)
#pragma unroll
      for (int r = 0; r < 2; ++r) {
        const int K = kc * 4 + half * 2 + r;      // row index (f*4 + J)
        const int f = K >> 2, J = K & 3;
        const float* vp = feat + (((size_t)f * LAT_V + (size_t)xn) * 12 + (size_t)(J * 3)) * 2;
        float vr[3], vi[3];
#pragma unroll
        for (int g = 0; g < 3; ++g) { vr[g] = vp[g * 2]; vi[g] = vp[g * 2 + 1]; }
#pragma unroll
        for (int i = 0; i < 3; ++i) {
          float tr, ti;
          if (ident) { tr = vr[i]; ti = vi[i]; }
          else {
            tr = fmaf(Mre[i][0], vr[0], fmaf(-Mim[i][0], vi[0],
                 fmaf(Mre[i][1], vr[1], fmaf(-Mim[i][1], vi[1],
                 fmaf(Mre[i][2], vr[2], -Mim[i][2] * vi[2])))));
            ti = fmaf(Mre[i][0], vi[0], fmaf(Mim[i][0], vr[0],
                 fmaf(Mre[i][1], vi[1], fmaf(Mim[i][1], vr[1],
                 fmaf(Mre[i][2], vi[2],  Mim[i][2] * vr[2])))));
          }
          b_re[i][kc][r] = tr;
          b_im[i][kc][r] = ti;
        }
      }

    // ---------- A fragments: spin weights W[(o,I)][(f,J)] for path p ----------
    // F32 WMMA has no A/B NEG modifier (NEG = CNeg only) -> keep -W_im fragment.
    v2f a_re[4], a_im[4], a_ng[4];
#pragma unroll
    for (int kc = 0; kc < 4; ++kc)
#pragma unroll
      for (int r = 0; r < 2; ++r) {
        const int K = kc * 4 + half * 2 + r;
        const int f = K >> 2, J = K & 3;
        const float* wp = W + (((((size_t)f * FOUT + oo) * NPATH + p) * 4 + II) * 4 + J) * 2;
        a_re[kc][r] =  wp[0];
        a_im[kc][r] =  wp[1];
        a_ng[kc][r] = -wp[1];
      }

    // ---------- complex 16x16x16 spin matmul: 4 real products, K chained in 4s ----------
#pragma unroll
    for (int g = 0; g < 3; ++g) {
#pragma unroll
      for (int kc = 0; kc < 4; ++kc) {
        acc_re[g] = __builtin_amdgcn_wmma_f32_16x16x4_f32(
            false, a_re[kc], false, b_re[g][kc], (short)0, acc_re[g], false, false);
        acc_re[g] = __builtin_amdgcn_wmma_f32_16x16x4_f32(
            false, a_ng[kc], false, b_im[g][kc], (short)0, acc_re[g], false, false);
        acc_im[g] = __builtin_amdgcn_wmma_f32_16x16x4_f32(
            false, a_re[kc], false, b_im[g][kc], (short)0, acc_im[g], false, false);
        acc_im[g] = __builtin_amdgcn_wmma_f32_16x16x4_f32(
            false, a_im[kc], false, b_re[g][kc], (short)0, acc_im[g], false, false);
      }
    }
  }

  // ---------- store D: C/D layout -> lanes 0-15 hold M=v, lanes 16-31 M=v+8 ----------
#pragma unroll
  for (int v = 0; v < 8; ++v) {
    const int M = v + half * 8;
    const int o = M >> 2, I = M & 3;
    float* op = out + (((size_t)o * LAT_V + (size_t)x) * 4 + (size_t)I) * 6;
#pragma unroll
    for (int g = 0; g < 3; ++g) {
      op[g * 2 + 0] = acc_re[g][v];
      op[g * 2 + 1] = acc_im[g][v];
    }
  }
}

extern "C" void kernel_launch(void* const* d_in, const int* in_sizes, int n_in,
                              void* d_out, int out_size, void* d_ws, size_t ws_size,
                              hipStream_t stream) {
  (void)in_sizes; (void)n_in; (void)out_size; (void)d_ws; (void)ws_size;
  const float* feat = (const float*)d_in[0];   // features_in, complex64 as float pairs
  const float* U    = (const float*)d_in[1];   // gauge links
  const float* W    = (const float*)d_in[2];   // weights
  float* out        = (float*)d_out;

  const int waves  = LAT_V / SITES_PER_WAVE;   // 16384 wave-tiles
  const int blocks = waves / WAVES_PER_BLK;    // 2048 blocks of 8 waves
  ptc5d_wmma_kernel<<<blocks, WAVES_PER_BLK * 32, 0, stream>>>(feat, U, W, out);
}